// GCN_51127290691793
// MI455X (gfx1250) — compile-verified
//
#include <hip/hip_runtime.h>
#include <math.h>

// ---------------------------------------------------------------------------
// GCN forward:  3x (GEMM via V_WMMA_F32_16X16X4_F32 + edge scatter-agg) + pool
// ---------------------------------------------------------------------------

#define N_NODES   50000
#define N_EDGES   1600000
#define N_GRAPHS  100
#define IN_FEAT   33
#define HIDDEN    128
#define N_CLASSES 2

typedef float v2f __attribute__((ext_vector_type(2)));
typedef float v8f __attribute__((ext_vector_type(8)));

// ---------------- degree / normalization ----------------
__global__ void k_fill1(float* __restrict__ p, int n) {
    int t = blockIdx.x * blockDim.x + threadIdx.x;
    if (t < n) p[t] = 1.0f;                       // self-loop contributes 1 to deg
}

__global__ void k_deg_edges(const int* __restrict__ dst, float* __restrict__ deg, int E) {
    int t = blockIdx.x * blockDim.x + threadIdx.x;
    if (t < E) atomicAdd(&deg[dst[t]], 1.0f);
}

__global__ void k_rsqrt_inplace(float* __restrict__ p, int n) {
    int t = blockIdx.x * blockDim.x + threadIdx.x;
    if (t < n) p[t] = rsqrtf(p[t]);               // deg >= 1 always (self-loops)
}

// ---------------- dense GEMM: Out[N x 128] = H[N x K] @ W[K x 128] ----------
// One wave per 16x16 output tile, fp32 WMMA (exact math vs reference).
// A 16x4 f32 layout: lanes 0-15 rows M=0..15, VGPR0 holds K=k0+2*hi,
// VGPR1 holds K=k0+1+2*hi (hi = lane>=16). B 4x16: same K split, column = lane&15.
// C/D: VGPR i -> row (i + 8*hi), col = lane&15.
template <int K, int LDH>
__global__ void k_gemm_wmma(const float* __restrict__ H,
                            const float* __restrict__ W,
                            float* __restrict__ Out) {
    const int lane   = threadIdx.x & 31;
    const int wid    = blockIdx.x * (blockDim.x >> 5) + (threadIdx.x >> 5);
    const int tile_m = wid >> 3;           // 8 n-tiles per m-tile (HIDDEN/16)
    const int tile_n = wid & 7;
    const int m      = lane & 15;
    const int hi     = lane >> 4;

    const float* __restrict__ Hr = H + (size_t)(tile_m * 16 + m) * LDH;
    const int col = tile_n * 16 + m;

    v8f acc = {0.f, 0.f, 0.f, 0.f, 0.f, 0.f, 0.f, 0.f};

    for (int k0 = 0; k0 < K; k0 += 4) {
        const int ka = k0 + 2 * hi;
        v2f a, b;
        a.x = (ka     < K) ? Hr[ka]     : 0.f;
        a.y = (ka + 1 < K) ? Hr[ka + 1] : 0.f;
        b.x = (ka     < K) ? W[(size_t)ka * HIDDEN + col]       : 0.f;
        b.y = (ka + 1 < K) ? W[(size_t)(ka + 1) * HIDDEN + col] : 0.f;
        acc = __builtin_amdgcn_wmma_f32_16x16x4_f32(
            /*neg_a=*/false, a, /*neg_b=*/false, b,
            /*c_mod=*/(short)0, acc, /*reuse_a=*/false, /*reuse_b=*/false);
    }

    float* __restrict__ Or = Out + (size_t)(tile_m * 16 + 8 * hi) * HIDDEN + col;
#pragma unroll
    for (int i = 0; i < 8; ++i) Or[(size_t)i * HIDDEN] = acc[i];
}

// ---------------- aggregation ------------------------------------------------
// out[i,:] = dinv[i]^2 * h[i,:]   (self-loop folded into the zero-init pass)
__global__ void k_self_init(const float* __restrict__ h, const float* __restrict__ dinv,
                            float* __restrict__ out, int n_elems) {
    int t = blockIdx.x * blockDim.x + threadIdx.x;
    if (t >= n_elems) return;
    const float w = dinv[t >> 7];              // t / HIDDEN
    out[t] = w * w * h[t];
}

// one wave per edge: gather h[src] (float4 per lane), scatter-add into out[dst]
__global__ void k_edge_agg(const int* __restrict__ src, const int* __restrict__ dst,
                           const float* __restrict__ dinv,
                           const float* __restrict__ h, float* __restrict__ out, int E) {
    int t = blockIdx.x * blockDim.x + threadIdx.x;
    int e = t >> 5;
    int lane = t & 31;
    if (e >= E) return;
    const int s = src[e];
    const int d = dst[e];
    const float w = dinv[s] * dinv[d];
    const float4 v = ((const float4*)(h + (size_t)s * HIDDEN))[lane];
    float* o = out + (size_t)d * HIDDEN + lane * 4;
    atomicAdd(o + 0, w * v.x);
    atomicAdd(o + 1, w * v.y);
    atomicAdd(o + 2, w * v.z);
    atomicAdd(o + 3, w * v.w);
}

template <bool RELU>
__global__ void k_bias_act(float* __restrict__ h, const float* __restrict__ b, int n_elems) {
    int t = blockIdx.x * blockDim.x + threadIdx.x;
    if (t >= n_elems) return;
    float v = h[t] + b[t & (HIDDEN - 1)];
    h[t] = RELU ? fmaxf(v, 0.f) : v;
}

// ---------------- pooling + final linear ------------------------------------
__global__ void k_zero(float* __restrict__ p, int n) {
    int t = blockIdx.x * blockDim.x + threadIdx.x;
    if (t < n) p[t] = 0.f;
}

__global__ void k_pool_acc(const float* __restrict__ h, const int* __restrict__ batch,
                           float* __restrict__ sums, int n_elems) {
    int t = blockIdx.x * blockDim.x + threadIdx.x;
    if (t >= n_elems) return;
    const int i = t >> 7;
    atomicAdd(&sums[(size_t)batch[i] * HIDDEN + (t & (HIDDEN - 1))], h[t]);
}

__global__ void k_pool_cnt(const int* __restrict__ batch, float* __restrict__ counts, int n) {
    int t = blockIdx.x * blockDim.x + threadIdx.x;
    if (t < n) atomicAdd(&counts[batch[t]], 1.0f);
}

__global__ void k_final_linear(const float* __restrict__ sums, const float* __restrict__ counts,
                               const float* __restrict__ Wlin, const float* __restrict__ blin,
                               float* __restrict__ out) {
    __shared__ float s0[HIDDEN];
    __shared__ float s1[HIDDEN];
    const int g = blockIdx.x;
    const int t = threadIdx.x;
    float c = counts[g];
    c = c > 1.0f ? c : 1.0f;
    const float p = sums[(size_t)g * HIDDEN + t] / c;
    s0[t] = p * Wlin[t * N_CLASSES + 0];
    s1[t] = p * Wlin[t * N_CLASSES + 1];
    __syncthreads();
    for (int off = HIDDEN / 2; off > 0; off >>= 1) {
        if (t < off) { s0[t] += s0[t + off]; s1[t] += s1[t + off]; }
        __syncthreads();
    }
    if (t == 0) {
        out[g * N_CLASSES + 0] = s0[0] + blin[0];
        out[g * N_CLASSES + 1] = s1[0] + blin[1];
    }
}

// ---------------------------------------------------------------------------
extern "C" void kernel_launch(void* const* d_in, const int* in_sizes, int n_in,
                              void* d_out, int out_size, void* d_ws, size_t ws_size,
                              hipStream_t stream) {
    const float* x     = (const float*)d_in[0];
    const int*   ei    = (const int*)d_in[1];     // (2, E): row0=src, row1=dst
    const int*   batch = (const int*)d_in[2];
    const float* W1    = (const float*)d_in[3];
    const float* b1    = (const float*)d_in[4];
    const float* W2    = (const float*)d_in[5];
    const float* b2    = (const float*)d_in[6];
    const float* W3    = (const float*)d_in[7];
    const float* b3    = (const float*)d_in[8];
    const float* Wlin  = (const float*)d_in[9];
    const float* blin  = (const float*)d_in[10];
    float* out = (float*)d_out;

    const int N = N_NODES, E = N_EDGES;
    const int* src = ei;
    const int* dst = ei + E;

    // workspace layout (floats), 256B-aligned chunks
    float* ws     = (float*)d_ws;
    float* dinv   = ws;                            // N
    float* bufA   = ws + 50048;                    // N*HIDDEN (GEMM output)
    float* bufB   = bufA + (size_t)N * HIDDEN;     // N*HIDDEN (aggregated)
    float* sums   = bufB + (size_t)N * HIDDEN;     // N_GRAPHS*HIDDEN
    float* counts = sums + N_GRAPHS * HIDDEN;      // N_GRAPHS

    const int T = 256;
    const int nb_nodes = (N + T - 1) / T;                    // 196
    const int nb_edges = (E + T - 1) / T;                    // 6250
    const int nb_feat  = ((size_t)N * HIDDEN + T - 1) / T;   // 25000
    const int nb_eagg  = ((size_t)E * 32 + T - 1) / T;       // 200000
    const int nb_gemm  = (N / 16) * (HIDDEN / 16) / (T / 32);// 3125

    // ---- normalization: deg -> dinv ----
    k_fill1<<<nb_nodes, T, 0, stream>>>(dinv, N);
    k_deg_edges<<<nb_edges, T, 0, stream>>>(dst, dinv, E);
    k_rsqrt_inplace<<<nb_nodes, T, 0, stream>>>(dinv, N);

    // ---- layer 1: K = 33 ----
    k_gemm_wmma<IN_FEAT, IN_FEAT><<<nb_gemm, T, 0, stream>>>(x, W1, bufA);
    k_self_init<<<nb_feat, T, 0, stream>>>(bufA, dinv, bufB, N * HIDDEN);
    k_edge_agg<<<nb_eagg, T, 0, stream>>>(src, dst, dinv, bufA, bufB, E);
    k_bias_act<true><<<nb_feat, T, 0, stream>>>(bufB, b1, N * HIDDEN);

    // ---- layer 2: K = 128 ----
    k_gemm_wmma<HIDDEN, HIDDEN><<<nb_gemm, T, 0, stream>>>(bufB, W2, bufA);
    k_self_init<<<nb_feat, T, 0, stream>>>(bufA, dinv, bufB, N * HIDDEN);
    k_edge_agg<<<nb_eagg, T, 0, stream>>>(src, dst, dinv, bufA, bufB, E);
    k_bias_act<true><<<nb_feat, T, 0, stream>>>(bufB, b2, N * HIDDEN);

    // ---- layer 3: K = 128, no relu ----
    k_gemm_wmma<HIDDEN, HIDDEN><<<nb_gemm, T, 0, stream>>>(bufB, W3, bufA);
    k_self_init<<<nb_feat, T, 0, stream>>>(bufA, dinv, bufB, N * HIDDEN);
    k_edge_agg<<<nb_eagg, T, 0, stream>>>(src, dst, dinv, bufA, bufB, E);
    k_bias_act<false><<<nb_feat, T, 0, stream>>>(bufB, b3, N * HIDDEN);

    // ---- mean pool + final linear ----
    k_zero<<<(N_GRAPHS * HIDDEN + N_GRAPHS + T - 1) / T, T, 0, stream>>>(sums, N_GRAPHS * HIDDEN + N_GRAPHS);
    k_pool_acc<<<nb_feat, T, 0, stream>>>(bufB, batch, sums, N * HIDDEN);
    k_pool_cnt<<<nb_nodes, T, 0, stream>>>(batch, counts, N);
    k_final_linear<<<N_GRAPHS, HIDDEN, 0, stream>>>(sums, counts, Wlin, blin, out);
}